// ObjectLoss_82386062672211
// MI455X (gfx1250) — compile-verified
//
#include <hip/hip_runtime.h>

#define NUM_PIDS 50000

typedef __attribute__((ext_vector_type(2))) float v2f;
typedef __attribute__((ext_vector_type(4))) float v4f;
typedef __attribute__((ext_vector_type(8))) float v8f;

// ---------------------------------------------------------------- zero scratch
__global__ void ObjectLoss_zero_ws(float* __restrict__ ws, int n) {
    int i = blockIdx.x * blockDim.x + threadIdx.x;
    if (i < n) ws[i] = 0.0f;
}

// --------------------------------------------------- mse + segmented scatter
// One wave32 processes a 16-row tile. Each lane loads row (tile*16 + lane&15)
// of pred/track (lanes 16..31 duplicate lanes 0..15's rows -> WGP$ hits, no
// extra HBM traffic). diff goes into the documented 16x4 f32 A-matrix layout;
// since we need diff * diff^T, the A and B register images are identical, so
// the same v2f feeds SRC0 and SRC1 of two chained v_wmma_f32_16x16x4_f32
// (K=0..3, K=4..7). Diagonal of the 16x16 accumulator = per-row mse; row m's
// value sits in lane m (m<8) / lane m+16 (m>=8), component = lane&7, and the
// owning lane's row index equals the row it already loaded. pid/rec loads are
// pinned at loop top with an empty asm so they issue in the same clause as the
// b128 streams instead of being sunk into the atomic branch (extra round-trip).
__global__ __launch_bounds__(256) void ObjectLoss_mse_seg(
    const float* __restrict__ pred, const float* __restrict__ track,
    const int* __restrict__ pid, const int* __restrict__ rec,
    float* __restrict__ seg_mse, float* __restrict__ seg_cnt, int n) {

    const int lane          = threadIdx.x & 31;
    const int waveInBlock   = threadIdx.x >> 5;
    const int wavesPerBlock = blockDim.x >> 5;
    const long waveId   = (long)blockIdx.x * wavesPerBlock + waveInBlock;
    const long numWaves = (long)gridDim.x * wavesPerBlock;

    const int  m      = lane & 15;              // row within tile this lane owns
    const bool hiHalf = lane >= 16;
    const bool activeLane = (lane < 8) || (lane >= 24);   // diagonal owners
    const long numTiles = ((long)n + 15) >> 4;

    for (long tile = waveId; tile < numTiles; tile += numWaves) {
        long row = tile * 16 + m;
        long r   = (row < n) ? row : (long)(n - 1);   // clamp for tail tile

        // --- issue ALL loads up front so they share a clause / overlap ------
        const float* pp = pred  + r * 8;
        const float* tp = track + r * 8;
        v4f p0 = __builtin_nontemporal_load((const v4f*)(pp));
        v4f p1 = __builtin_nontemporal_load((const v4f*)(pp + 4));
        v4f t0 = __builtin_nontemporal_load((const v4f*)(tp));
        v4f t1 = __builtin_nontemporal_load((const v4f*)(tp + 4));
        int  pv = pid[r];
        int  rv = rec[r];
        // Anchor: force pv/rv to materialize here so the loads cannot be sunk
        // into the (divergent) atomic branch after the WMMAs.
        asm volatile("" : "+v"(pv), "+v"(rv));

        v4f d0 = p0 - t0;   // diff[0..3]
        v4f d1 = p1 - t1;   // diff[4..7]
        if (row >= (long)n) {                          // keep WMMA inputs defined
            v4f z = {0.f, 0.f, 0.f, 0.f};
            d0 = z; d1 = z;
        }

        // A-matrix (16x4 f32): VGPR0 = K0 (lanes 0-15) / K2 (lanes 16-31),
        //                      VGPR1 = K1 / K3.  B = A^T has the same image.
        v2f A, B;
        A.x = hiHalf ? d0.z : d0.x;   A.y = hiHalf ? d0.w : d0.y;   // K=0..3
        B.x = hiHalf ? d1.z : d1.x;   B.y = hiHalf ? d1.w : d1.y;   // K=4..7

        v8f acc = {};
        acc = __builtin_amdgcn_wmma_f32_16x16x4_f32(false, A, false, A,
                                                    (short)0, acc, false, false);
        acc = __builtin_amdgcn_wmma_f32_16x16x4_f32(false, B, false, B,
                                                    (short)0, acc, false, false);

        // Diagonal extraction: component index is lane&7 for both owner groups.
        const int comp = lane & 7;
        float mse = acc[0];
        #pragma unroll
        for (int c = 1; c < 8; ++c) mse = (comp == c) ? acc[c] : mse;

        // Single predicated atomic pair; pv/rv already resident (prefetched).
        if (activeLane && row < (long)n && pv > 0 && pv < NUM_PIDS && rv > 0) {
            atomicAdd(&seg_mse[pv], mse);
            atomicAdd(&seg_cnt[pv], 1.0f);
        }
    }
}

// ------------------------------------------------- per-segment mean + count K
__global__ __launch_bounds__(256) void ObjectLoss_finalize(
    const float* __restrict__ seg_mse, const float* __restrict__ seg_cnt,
    float* __restrict__ accum /* [0]=sum of means, [1]=K */) {
    __shared__ float s_sum[256];
    __shared__ float s_k[256];
    const int tid = threadIdx.x;

    float sum = 0.f, k = 0.f;
    for (int p = blockIdx.x * blockDim.x + tid; p < NUM_PIDS;
         p += gridDim.x * blockDim.x) {
        float c = seg_cnt[p];
        if (c > 0.f) { sum += seg_mse[p] / c; k += 1.f; }
    }
    s_sum[tid] = sum; s_k[tid] = k;
    __syncthreads();
    for (int off = blockDim.x >> 1; off > 0; off >>= 1) {
        if (tid < off) { s_sum[tid] += s_sum[tid + off]; s_k[tid] += s_k[tid + off]; }
        __syncthreads();
    }
    if (tid == 0) {
        atomicAdd(&accum[0], s_sum[0]);
        atomicAdd(&accum[1], s_k[0]);
    }
}

__global__ void ObjectLoss_write_out(const float* __restrict__ accum,
                                     float* __restrict__ out) {
    if (threadIdx.x == 0 && blockIdx.x == 0) {
        float k = accum[1];
        out[0] = (k > 0.f) ? (100.f * accum[0] / k) : 0.f;
    }
}

// ---------------------------------------------------------------------- launch
extern "C" void kernel_launch(void* const* d_in, const int* in_sizes, int n_in,
                              void* d_out, int out_size, void* d_ws, size_t ws_size,
                              hipStream_t stream) {
    // setup_inputs order: 0=W 1=beta 2=H 3=pred 4=Y 5=particle_id 6=track_params 7=reconstructable
    const float* pred  = (const float*)d_in[3];
    const int*   pid   = (const int*)  d_in[5];
    const float* track = (const float*)d_in[6];
    const int*   rec   = (const int*)  d_in[7];
    const int    n     = in_sizes[5];            // N rows

    float* ws      = (float*)d_ws;
    float* seg_mse = ws;
    float* seg_cnt = ws + NUM_PIDS;
    float* accum   = ws + 2 * NUM_PIDS;          // [sum_of_means, K]

    const int zn = 2 * NUM_PIDS + 2;
    ObjectLoss_zero_ws<<<(zn + 255) / 256, 256, 0, stream>>>(ws, zn);

    // 2048 blocks x 8 waves = 16K waves grid-striding 125K 16-row tiles.
    ObjectLoss_mse_seg<<<2048, 256, 0, stream>>>(pred, track, pid, rec,
                                                 seg_mse, seg_cnt, n);

    ObjectLoss_finalize<<<128, 256, 0, stream>>>(seg_mse, seg_cnt, accum);
    ObjectLoss_write_out<<<1, 64, 0, stream>>>(accum, (float*)d_out);
}